// LogOddsPerformanceTransformer_23287312679081
// MI455X (gfx1250) — compile-verified
//
#include <hip/hip_runtime.h>
#include <math.h>

#define NBINS_MAX 4096
#define BLOCK 256

// ---- CDNA5 async global->LDS staging (probe-verified on this toolchain) ---
#if defined(__has_builtin)
#  if __has_builtin(__builtin_amdgcn_global_load_async_to_lds_b128) && \
      __has_builtin(__builtin_amdgcn_s_wait_asynccnt)
#    define USE_ASYNC_LDS 1
#  endif
#endif

typedef int   v4i __attribute__((vector_size(16)));
typedef float v4f __attribute__((ext_vector_type(4)));   // native clang vector
typedef __attribute__((address_space(1))) v4i gv4i_t;    // global (AS1)
typedef __attribute__((address_space(3))) v4i lv4i_t;    // LDS    (AS3)

// Map score -> largest bin edge <= score (searchsorted 'right' - 1, clipped),
// using uniform-spacing estimate + +/-1 fixup against the real bin values.
__device__ __forceinline__ float snap_to_bin(float x, const float* __restrict__ sbins,
                                             float binMin, float invStep, int nbins) {
  // logit; one accurate log instead of two
  float s = logf(x / (1.0f - x));
  s = fmaxf(s, binMin);                       // clamp below first edge
  float t = (s - binMin) * invStep;           // t >= 0
  int idx = (int)t;                           // trunc == floor for t >= 0
  if (idx > nbins - 1) idx = nbins - 1;
  // fixup: exact searchsorted semantics despite fp rounding in linspace edges
  if (idx > 0 && sbins[idx] > s) idx--;
  if (idx < nbins - 1 && sbins[idx + 1] <= s) idx++;
  return sbins[idx];
}

__device__ __forceinline__ v4f snap4(v4f x, const float* __restrict__ sbins,
                                     float binMin, float invStep, int nbins) {
  v4f r;
  r.x = snap_to_bin(x.x, sbins, binMin, invStep, nbins);
  r.y = snap_to_bin(x.y, sbins, binMin, invStep, nbins);
  r.z = snap_to_bin(x.z, sbins, binMin, invStep, nbins);
  r.w = snap_to_bin(x.w, sbins, binMin, invStep, nbins);
  return r;
}

__global__ __launch_bounds__(BLOCK) void logodds_bin_kernel(
    const float* __restrict__ Xs, const float* __restrict__ bins,
    float* __restrict__ out, int n, int nbins) {
  __shared__ float sbins[NBINS_MAX];

  // ---- stage bins (16 KB) into LDS via async DMA path -------------------
#if defined(USE_ASYNC_LDS)
  {
    gv4i_t* g = (gv4i_t*)bins;
    lv4i_t* l = (lv4i_t*)sbins;
    int nvec = nbins >> 2;
    for (int i = (int)threadIdx.x; i < nvec; i += BLOCK) {
      __builtin_amdgcn_global_load_async_to_lds_b128(g + i, l + i, 0, 0);
    }
    __builtin_amdgcn_s_wait_asynccnt(0);
  }
#else
  {
    const v4f* b4 = reinterpret_cast<const v4f*>(bins);
    v4f* s4 = reinterpret_cast<v4f*>(sbins);
    for (int i = threadIdx.x; i < (nbins >> 2); i += BLOCK) s4[i] = b4[i];
  }
#endif
  __syncthreads();

  const float binMin = sbins[0];
  const float binMax = sbins[nbins - 1];
  const float invStep = (float)(nbins - 1) / (binMax - binMin);

  // ---- main stream: non-temporal b128 loads/stores, 2x v4f / iter -------
  const int n4 = n >> 2;
  const v4f* X4 = reinterpret_cast<const v4f*>(Xs);
  v4f* O4 = reinterpret_cast<v4f*>(out);
  const int tid = (int)(blockIdx.x * blockDim.x + threadIdx.x);
  const int stride = (int)(gridDim.x * blockDim.x);

  int i = tid;
  for (; i + stride < n4; i += 2 * stride) {
    v4f x0 = __builtin_nontemporal_load(&X4[i]);
    v4f x1 = __builtin_nontemporal_load(&X4[i + stride]);
    v4f r0 = snap4(x0, sbins, binMin, invStep, nbins);
    v4f r1 = snap4(x1, sbins, binMin, invStep, nbins);
    __builtin_nontemporal_store(r0, &O4[i]);
    __builtin_nontemporal_store(r1, &O4[i + stride]);
  }
  for (; i < n4; i += stride) {
    v4f x = __builtin_nontemporal_load(&X4[i]);
    v4f r = snap4(x, sbins, binMin, invStep, nbins);
    __builtin_nontemporal_store(r, &O4[i]);
  }

  // tail (n % 4)
  for (int j = (n4 << 2) + tid; j < n; j += stride) {
    out[j] = snap_to_bin(Xs[j], sbins, binMin, invStep, nbins);
  }
}

extern "C" void kernel_launch(void* const* d_in, const int* in_sizes, int n_in,
                              void* d_out, int out_size, void* d_ws, size_t ws_size,
                              hipStream_t stream) {
  const float* Xs = (const float*)d_in[0];
  const float* bins = (const float*)d_in[1];
  float* out = (float*)d_out;
  int n = in_sizes[0];
  int nbins = (n_in > 1) ? in_sizes[1] : NBINS_MAX;
  if (nbins > NBINS_MAX) nbins = NBINS_MAX;

  int n4 = n >> 2;
  int grid = (n4 + BLOCK - 1) / BLOCK;
  if (grid > 4096) grid = 4096;
  if (grid < 1) grid = 1;

  logodds_bin_kernel<<<grid, BLOCK, 0, stream>>>(Xs, bins, out, n, nbins);
}